// gconv_37847251812930
// MI455X (gfx1250) — compile-verified
//
#include <hip/hip_runtime.h>
#include <hip/hip_bf16.h>

typedef __attribute__((ext_vector_type(2))) float v2f;
typedef __attribute__((ext_vector_type(4))) float v4f;
typedef __attribute__((ext_vector_type(8))) float v8f;
typedef __attribute__((ext_vector_type(4))) int   v4i;

// CDNA5 direct global->LDS async copy (ASYNCcnt-tracked), if the toolchain
// exposes the gfx1250 builtin. Fallback: register-staged double buffering.
#if __has_builtin(__builtin_amdgcn_global_load_async_to_lds_b128)
#define GCONV_ASYNC_LDS 1
#endif

__device__ __forceinline__ void gconv_wait_async() {
#if __has_builtin(__builtin_amdgcn_s_wait_asynccnt)
    __builtin_amdgcn_s_wait_asynccnt(0);
#elif defined(__AMDGCN__)
    asm volatile("s_wait_asynccnt 0x0" ::: "memory");
#endif
}

#if defined(GCONV_ASYNC_LDS)
// The builtin takes int4 pointers qualified with explicit address spaces:
//   arg0: int4 addrspace(1)* (global source)
//   arg1: int4 addrspace(3)* (LDS destination)
//   arg2/3: immediate offset / cache policy
typedef v4i __attribute__((address_space(1))) as1_v4i;
typedef v4i __attribute__((address_space(3))) as3_v4i;

// Copy 16B per lane, global -> LDS, asynchronously (global_load_async_to_lds_b128).
__device__ __forceinline__ void gconv_async_b128(const float* g, float* l) {
    __builtin_amdgcn_global_load_async_to_lds_b128(
        (as1_v4i*)(uintptr_t)g,
        (as3_v4i*)(unsigned int)(uintptr_t)l,  // LDS offset = low 32 bits
        0, 0);
}
#endif

// ---------------------------------------------------------------------------
// Generic batched GEMM:  C[z] (+)= A[z] @ B[z]  (+ bias)
//   A: row-major [M,K]  (lda, batch stride batchA; 0 => shared across batch)
//   B: row-major [K,Nc] (ldb, batch stride batchB)
//   C: row-major [M,Nc] (ldc, batch stride batchC)
// One workgroup computes a 64x64 tile of C for one batch element.
// 256 threads = 8 wave32s in a 4(M) x 2(N) arrangement; each wave computes a
// 16x32 slab via two f32 WMMA accumulators, K stepped by 16 (4 x K=4 WMMA).
// ---------------------------------------------------------------------------
__global__ __launch_bounds__(256) void gemm64_wmma_f32(
    const float* __restrict__ A, long batchA, int lda,
    const float* __restrict__ Bm, long batchB, int ldb,
    float* __restrict__ C, long batchC, int ldc,
    int K, int accum, const float* __restrict__ bias)
{
    // LDS tiles, double buffered. A padded to 20 floats/row:
    //  - row stride 80B keeps B128 stores 16B-aligned,
    //  - stride-20 column reads are bank-conflict-free across 16 lanes.
    __shared__ float lA[2][64][20];
    __shared__ float lB[2][16][64];

    const int t = threadIdx.x;
    const int L = t & 31;
    const int w = t >> 5;
    const int waveM = (w >> 1) << 4;   // 0,16,32,48
    const int waveN = (w & 1) << 5;    // 0,32

    const int z = blockIdx.z;
    const long m0 = (long)blockIdx.x * 64;
    const long n0 = (long)blockIdx.y * 64;

    const float* Ab = A + (long)z * batchA;
    const float* Bb = Bm + (long)z * batchB;
    float*       Cb = C + (long)z * batchC;

    // cooperative tile-load indices (one 16B transfer per thread per tile)
    const int la_r = t >> 2;            // 0..63
    const int la_q = (t & 3) << 2;      // 0,4,8,12
    const int lb_r = t >> 4;            // 0..15
    const int lb_q = (t & 15) << 2;     // 0,4,...,60

    // WMMA fragment addressing (ISA 16x16x4 f32 layouts, wave32)
    const int arow = waveM + (L & 15);  // A-fragment row
    const int kb   = (L >> 4) << 1;     // K pair base: lanes 0-15 -> 0, 16-31 -> 2
    const int bcol = waveN + (L & 15);  // B/C column within tile

    v8f acc0 = {};
    v8f acc1 = {};

    const int nk = K >> 4;

#if defined(GCONV_ASYNC_LDS)
    // prologue: DMA k-tile 0 into buffer 0 (no VGPR staging)
    gconv_async_b128(Ab + (m0 + la_r) * (long)lda + la_q, &lA[0][la_r][la_q]);
    gconv_async_b128(Bb + (long)lb_r * ldb + n0 + lb_q, &lB[0][lb_r][lb_q]);
    gconv_wait_async();
    __syncthreads();

    for (int ks = 0; ks < nk; ++ks) {
        const int cur = ks & 1;
        const int nxt = cur ^ 1;
        const int k0n = (ks + 1) << 4;

        if (ks + 1 < nk) {  // fire next tile's DMA; overlaps with compute below
            gconv_async_b128(Ab + (m0 + la_r) * (long)lda + k0n + la_q,
                             &lA[nxt][la_r][la_q]);
            gconv_async_b128(Bb + (long)(k0n + lb_r) * ldb + n0 + lb_q,
                             &lB[nxt][lb_r][lb_q]);
        }

#pragma unroll
        for (int kk = 0; kk < 16; kk += 4) {
            v2f a;
            a.x = lA[cur][arow][kk + kb];
            a.y = lA[cur][arow][kk + kb + 1];
            v2f b0;
            b0.x = lB[cur][kk + kb][bcol];
            b0.y = lB[cur][kk + kb + 1][bcol];
            v2f b1;
            b1.x = lB[cur][kk + kb][bcol + 16];
            b1.y = lB[cur][kk + kb + 1][bcol + 16];
            acc0 = __builtin_amdgcn_wmma_f32_16x16x4_f32(
                false, a, false, b0, (short)0, acc0, false, false);
            acc1 = __builtin_amdgcn_wmma_f32_16x16x4_f32(
                false, a, false, b1, (short)0, acc1, false, false);
        }

        gconv_wait_async();   // this wave's DMA into nxt has landed
        __syncthreads();      // all waves' DMAs have landed
    }
#else
    // fallback: register-staged double buffering
    v4f rA = *(const v4f*)(Ab + (m0 + la_r) * (long)lda + la_q);
    v4f rB = *(const v4f*)(Bb + (long)lb_r * ldb + n0 + lb_q);
    *(v4f*)&lA[0][la_r][la_q] = rA;
    *(v4f*)&lB[0][lb_r][lb_q] = rB;
    __syncthreads();

    for (int ks = 0; ks < nk; ++ks) {
        const int cur = ks & 1;
        const int nxt = cur ^ 1;
        const int k0n = (ks + 1) << 4;

        if (ks + 1 < nk) {
            rA = *(const v4f*)(Ab + (m0 + la_r) * (long)lda + k0n + la_q);
            rB = *(const v4f*)(Bb + (long)(k0n + lb_r) * ldb + n0 + lb_q);
        }

#pragma unroll
        for (int kk = 0; kk < 16; kk += 4) {
            v2f a;
            a.x = lA[cur][arow][kk + kb];
            a.y = lA[cur][arow][kk + kb + 1];
            v2f b0;
            b0.x = lB[cur][kk + kb][bcol];
            b0.y = lB[cur][kk + kb + 1][bcol];
            v2f b1;
            b1.x = lB[cur][kk + kb][bcol + 16];
            b1.y = lB[cur][kk + kb + 1][bcol + 16];
            acc0 = __builtin_amdgcn_wmma_f32_16x16x4_f32(
                false, a, false, b0, (short)0, acc0, false, false);
            acc1 = __builtin_amdgcn_wmma_f32_16x16x4_f32(
                false, a, false, b1, (short)0, acc1, false, false);
        }

        if (ks + 1 < nk) {
            *(v4f*)&lA[nxt][la_r][la_q] = rA;
            *(v4f*)&lB[nxt][lb_r][lb_q] = rB;
        }
        __syncthreads();
    }
#endif

    // epilogue: C VGPR r holds row (r + (L>=16 ? 8 : 0)), col = L&15
    const int rofs = (L >> 4) << 3;
    float bias0 = 0.0f, bias1 = 0.0f;
    if (bias) {
        bias0 = bias[n0 + bcol];
        bias1 = bias[n0 + bcol + 16];
    }
#pragma unroll
    for (int r = 0; r < 8; ++r) {
        const long row = m0 + waveM + r + rofs;
        const long i0 = row * (long)ldc + n0 + bcol;
        const long i1 = i0 + 16;
        float v0 = acc0[r] + bias0;
        float v1 = acc1[r] + bias1;
        if (accum) { v0 += Cb[i0]; v1 += Cb[i1]; }
        Cb[i0] = v0;
        Cb[i1] = v1;
    }
}

// Wt[d][h] = W[h][d]  (W: [64,448] row-major -> Wt: [448,64] row-major)
__global__ __launch_bounds__(256) void transpose_w(
    const float* __restrict__ W, float* __restrict__ Wt)
{
    const int i = blockIdx.x * 256 + threadIdx.x;
    if (i < 448 * 64) {
        const int d = i >> 6;
        const int h = i & 63;
        Wt[i] = W[h * 448 + d];
    }
}

// ---------------------------------------------------------------------------
// out = cat([x, score@(a0@x), score@(a0^2@x), score@(a1@x), score@(a1^2@x),
//            score@(adj@x), score@(adj^2@x)]) @ W^T + b
// Restructured:  out = x @ W1^T + score @ P + b,  P = sum_c y_c @ Wc^T
// (score is applied exactly once; concat never materialized)
// ---------------------------------------------------------------------------
extern "C" void kernel_launch(void* const* d_in, const int* in_sizes, int n_in,
                              void* d_out, int out_size, void* d_ws, size_t ws_size,
                              hipStream_t stream)
{
    const float* x     = (const float*)d_in[0];  // [128,512,64]
    const float* adj   = (const float*)d_in[1];  // [512,512]
    const float* score = (const float*)d_in[2];  // [128,512,512]
    const float* adj0  = (const float*)d_in[3];  // [512,512]
    const float* adj1  = (const float*)d_in[4];  // [512,512]
    const float* W     = (const float*)d_in[5];  // [64,448]
    const float* bias  = (const float*)d_in[6];  // [64]
    float* out = (float*)d_out;                  // [128,512,64]
    float* ws  = (float*)d_ws;

    const int B = 128, N = 512, H = 64;
    const long sX = (long)N * H;      // 32768 : batch stride of [B,N,H]
    const long sS = (long)N * N;      // 262144: batch stride of score

    float* Wt   = ws;                          // 448*64
    float* buf1 = Wt + 448 * 64;               // [B,N,H]
    float* buf2 = buf1 + (long)B * N * H;      // [B,N,H]
    float* P    = buf2 + (long)B * N * H;      // [B,N,H]

    transpose_w<<<112, 256, 0, stream>>>(W, Wt);

    dim3 blk(256);
    dim3 grd(N / 64, 1, B);  // (8,1,128): 64x64 C tiles, Nc = 64

    const float* sup[3] = {adj0, adj1, adj};
    for (int s = 0; s < 3; ++s) {
        // y1 = a_s @ x          (A shared across batch)
        gemm64_wmma_f32<<<grd, blk, 0, stream>>>(
            sup[s], 0, N, x, sX, H, buf1, sX, H, N, /*accum=*/0, nullptr);
        // P (+)= y1 @ Wc^T,  chunk c = 2s  -> Wt rows [64+128s, 128+128s)
        gemm64_wmma_f32<<<grd, blk, 0, stream>>>(
            buf1, sX, H, Wt + (long)(64 + 128 * s) * 64, 0, H,
            P, sX, H, H, /*accum=*/(s == 0) ? 0 : 1, nullptr);
        // y2 = a_s @ y1
        gemm64_wmma_f32<<<grd, blk, 0, stream>>>(
            sup[s], 0, N, buf1, sX, H, buf2, sX, H, N, 0, nullptr);
        // P += y2 @ Wc^T,  chunk c = 2s+1 -> Wt rows [128+128s, 192+128s)
        gemm64_wmma_f32<<<grd, blk, 0, stream>>>(
            buf2, sX, H, Wt + (long)(128 + 128 * s) * 64, 0, H,
            P, sX, H, H, 1, nullptr);
    }

    // out = x @ W[:, :64]^T + bias
    gemm64_wmma_f32<<<grd, blk, 0, stream>>>(
        x, sX, H, Wt, 0, H, out, sX, H, H, 0, bias);

    // out += score @ P   (score read exactly once)
    gemm64_wmma_f32<<<grd, blk, 0, stream>>>(
        score, sS, N, P, sX, H, out, sX, H, N, 1, nullptr);
}